// YOLOLayer_35536559407824
// MI455X (gfx1250) — compile-verified
//
#include <hip/hip_runtime.h>

// YOLO head decode: [N, A*(C+5), H, W] -> [N, A*H*W, C+5] with fused activations.
// ~418 MB traffic -> ~18 us floor @ 23.3 TB/s. One block per (n, a, y):
//   Phase 1: b128 NT global loads (coalesced rows), activation in registers
//            (channel uniform per chunk), transpose-scatter b32 to LDS in
//            OUTPUT layout [col*85 + c] (2-way bank conflict max).
//   Phase 2: contiguous LDS -> global via CDNA5 GLOBAL_STORE_ASYNC_FROM_LDS_B128.

#define NB      32
#define AA      3
#define HH      80
#define WW      80
#define CH      85             // C + 5
#define HW      (HH * WW)      // 6400
#define IMG2    409600.0f      // 640 * 640 (reference double-scales)
#define TILE    (CH * WW)      // 6800 floats per block (contiguous output span)
#define CHUNKS  (TILE / 4)     // 1700 float4 chunks
#define THREADS 256
#define LOG2E   1.44269504088896340736f

#if defined(__has_builtin)
#if __has_builtin(__builtin_amdgcn_global_store_async_from_lds_b128)
#define USE_ASYNC_ST128 1
#endif
#endif

typedef float __attribute__((ext_vector_type(4))) f32x4;
typedef int   __attribute__((ext_vector_type(4))) i32x4;
typedef __attribute__((address_space(1))) i32x4 glb_i4;
typedef __attribute__((address_space(3))) i32x4 lds_i4;
typedef __attribute__((address_space(1))) float glb_f;
typedef __attribute__((address_space(3))) float lds_f;

__global__ __launch_bounds__(THREADS) void yolo_decode_kernel(
    const float* __restrict__ x, float* __restrict__ out) {
  __shared__ __align__(16) float tile[TILE];  // 27200 B, activated, output layout

  const int bid = blockIdx.x;         // bid = n*240 + a*80 + y
  const int y   = bid % HH;
  const int na  = bid / HH;
  const int a   = na % AA;
  const int n   = na / AA;
  const int tid = threadIdx.x;

  // input flat index of (n, a*85+c, y, col) = in_base + c*6400 + col
  const long long in_base =
      (long long)n * (AA * CH * HW) + (long long)a * (CH * HW) + (long long)y * WW;
  const float* src = x + in_base;

  const float aw2 = ((a == 0) ? 10.0f : (a == 1) ? 16.0f : 33.0f) * IMG2;
  const float ah2 = ((a == 0) ? 13.0f : (a == 1) ? 30.0f : 23.0f) * IMG2;
  const float fy  = (float)y;

  // ---- Phase 1: load row chunks (b128, NT), activate, transpose-scatter to LDS ----
  for (int q = tid; q < CHUNKS; q += THREADS) {
    const int c    = q / 20;        // channel row: uniform across this chunk's 4 elems
    const int col4 = q - c * 20;    // float4 index within the 80-wide row
    const f32x4 v4 =
        __builtin_nontemporal_load((const f32x4*)(src + c * HW) + col4);
    const bool is_wh = (c == 2) | (c == 3);
    float* lp = &tile[(col4 * 4) * CH + c];  // output layout: [col*85 + c]
#pragma unroll
    for (int k = 0; k < 4; ++k) {
      const float v   = v4[k];
      const float t   = __builtin_amdgcn_exp2f((is_wh ? v : -v) * LOG2E);
      const float sig = __builtin_amdgcn_rcpf(1.0f + t);  // 1/(1+exp(-v))
      float res;
      if (c >= 4)      res = sig;                                   // obj / class
      else if (c == 0) res = (sig + (float)(col4 * 4 + k)) * IMG2;  // x + grid_x
      else if (c == 1) res = (sig + fy) * IMG2;                     // y + grid_y
      else if (c == 2) res = t * aw2;                               // w * anchor_w
      else             res = t * ah2;                               // h * anchor_h
      lp[k * CH] = res;
    }
  }
  __syncthreads();

  float* dst = out + (long long)bid * TILE;  // contiguous 6800-float output span

  // ---- Phase 2: contiguous LDS -> global copy, b128 per lane ----
  for (int j = tid; j < CHUNKS; j += THREADS) {
#ifdef USE_ASYNC_ST128
    // CDNA5 async DMA: LDS -> global without touching VGPRs (ASYNCcnt tracked)
    __builtin_amdgcn_global_store_async_from_lds_b128(
        (glb_i4*)(glb_f*)dst + j, (lds_i4*)(lds_f*)tile + j, 0, 0);
#else
    const f32x4 vv = *((const f32x4*)tile + j);
    __builtin_nontemporal_store(vv, (f32x4*)dst + j);
#endif
  }
#ifdef USE_ASYNC_ST128
#if __has_builtin(__builtin_amdgcn_s_wait_asynccnt)
  __builtin_amdgcn_s_wait_asynccnt(0);
#else
  asm volatile("s_wait_asynccnt 0" ::: "memory");
#endif
#endif
}

extern "C" void kernel_launch(void* const* d_in, const int* in_sizes, int n_in,
                              void* d_out, int out_size, void* d_ws, size_t ws_size,
                              hipStream_t stream) {
  (void)in_sizes; (void)n_in; (void)out_size; (void)d_ws; (void)ws_size;
  const float* x = (const float*)d_in[0];
  float* out     = (float*)d_out;
  dim3 grid(NB * AA * HH);  // 7680 blocks, one per (n, a, y)
  dim3 block(THREADS);      // 8 waves (wave32)
  yolo_decode_kernel<<<grid, block, 0, stream>>>(x, out);
}